// GATLayer_6691559047407
// MI455X (gfx1250) — compile-verified
//
#include <hip/hip_runtime.h>

#define N_NODES 50000
#define D 128
#define NEG_SLOPE 0.2f

typedef __attribute__((ext_vector_type(2))) float v2f;
typedef __attribute__((ext_vector_type(4))) float v4f;
typedef __attribute__((ext_vector_type(8))) float v8f;

__device__ __forceinline__ float leaky(float e) {
    return e > 0.f ? e : NEG_SLOPE * e;
}

// ---------------------------------------------------------------------------
// H[N,D] = X[N,D] @ W[D,D] using V_WMMA_F32_16X16X4_F32.
// Block = 256 threads = 8 waves. Block owns one 16-row panel; wave w owns the
// 16x16 output tile at columns [16w, 16w+16). W (64KB) and the X panel (8KB)
// are staged in LDS.
// ---------------------------------------------------------------------------
__global__ __launch_bounds__(256) void k_gemm(const float* __restrict__ X,
                                              const float* __restrict__ W,
                                              float* __restrict__ H) {
    __shared__ float Ws[D * D];    // 64 KB
    __shared__ float Xs[16 * D];   // 8 KB
    const int tid  = threadIdx.x;
    const int lane = tid & 31;
    const int wave = tid >> 5;
    const int m0   = blockIdx.x * 16;

    // Cooperative float4 loads into LDS
    const v4f* Wv  = (const v4f*)W;
    v4f*       Wsv = (v4f*)Ws;
#pragma unroll
    for (int i = 0; i < (D * D / 4) / 256; ++i)   // 16 iters
        Wsv[tid + i * 256] = Wv[tid + i * 256];

    const v4f* Xv  = (const v4f*)(X + (size_t)m0 * D);
    v4f*       Xsv = (v4f*)Xs;
#pragma unroll
    for (int i = 0; i < (16 * D / 4) / 256; ++i)  // 2 iters
        Xsv[tid + i * 256] = Xv[tid + i * 256];

    __syncthreads();

    const int half    = lane >> 4;   // 0: K+{0,1}, 1: K+{2,3}
    const int l16     = lane & 15;
    const int colbase = wave * 16;

    v8f c = {};
#pragma unroll
    for (int k0 = 0; k0 < D; k0 += 4) {
        const int kk = k0 + half * 2;
        v2f a, b;
        // A 16x4: lane row = l16, two consecutive K per lane-half
        a.x = Xs[l16 * D + kk];
        a.y = Xs[l16 * D + kk + 1];
        // B 4x16: row K striped across lanes within a VGPR
        b.x = Ws[kk * D + colbase + l16];
        b.y = Ws[(kk + 1) * D + colbase + l16];
        c = __builtin_amdgcn_wmma_f32_16x16x4_f32(
            /*neg_a=*/false, a, /*neg_b=*/false, b,
            /*c_mod=*/(short)0, c, /*reuse_a=*/false, /*reuse_b=*/false);
    }

    // C/D layout: VGPR r -> M = r + 8*half, N = l16
#pragma unroll
    for (int r = 0; r < 8; ++r) {
        const int row = m0 + r + half * 8;
        H[(size_t)row * D + colbase + l16] = c[r];
    }
}

// ---------------------------------------------------------------------------
// Per-node attention scalars: as[i] = h_i . a_src ; ad[i] = h_i . a_dst.
// One wave per node, lane handles a float4 slice, shuffle reduction.
// Also seeds the segment max with the self-loop score.
// ---------------------------------------------------------------------------
__global__ __launch_bounds__(256) void k_alpha(const float* __restrict__ H,
                                               const float* __restrict__ a_src,
                                               const float* __restrict__ a_dst,
                                               float* __restrict__ as,
                                               float* __restrict__ ad,
                                               float* __restrict__ m) {
    const int lane = threadIdx.x & 31;
    const int node = (int)((blockIdx.x * blockDim.x + threadIdx.x) >> 5);
    if (node >= N_NODES) return;
    v4f h  = *(const v4f*)(H + (size_t)node * D + lane * 4);
    v4f vs = *(const v4f*)(a_src + lane * 4);
    v4f vd = *(const v4f*)(a_dst + lane * 4);
    float s = h.x * vs.x + h.y * vs.y + h.z * vs.z + h.w * vs.w;
    float d = h.x * vd.x + h.y * vd.y + h.z * vd.z + h.w * vd.w;
#pragma unroll
    for (int off = 16; off > 0; off >>= 1) {
        s += __shfl_xor(s, off, 32);
        d += __shfl_xor(d, off, 32);
    }
    if (lane == 0) {
        as[node] = s;
        ad[node] = d;
        m[node]  = leaky(s + d);   // self-loop always present
    }
}

// Pass 1: segment max over incoming edges (float atomic max -> v_max_num path)
__global__ void k_edge_max(const int* __restrict__ src, const int* __restrict__ dst,
                           const float* __restrict__ as, const float* __restrict__ ad,
                           float* __restrict__ m, int E) {
    int e = blockIdx.x * blockDim.x + threadIdx.x;
    if (e >= E) return;
    int s = src[e], d = dst[e];
    float v = leaky(as[s] + ad[d]);
    atomicMax(&m[d], v);
}

// Seed denominator with the self-loop term
__global__ void k_denom_init(const float* __restrict__ as, const float* __restrict__ ad,
                             const float* __restrict__ m, float* __restrict__ dn) {
    int i = blockIdx.x * blockDim.x + threadIdx.x;
    if (i >= N_NODES) return;
    dn[i] = __expf(leaky(as[i] + ad[i]) - m[i]);
}

// Pass 2: denom[dst] += exp(e - m[dst])
__global__ void k_edge_denom(const int* __restrict__ src, const int* __restrict__ dst,
                             const float* __restrict__ as, const float* __restrict__ ad,
                             const float* __restrict__ m, float* __restrict__ dn, int E) {
    int e = blockIdx.x * blockDim.x + threadIdx.x;
    if (e >= E) return;
    int s = src[e], d = dst[e];
    float v = leaky(as[s] + ad[d]);
    atomicAdd(&dn[d], __expf(v - m[d]));
}

// Init output with the self-loop contribution: Y[i] = alpha_self * h[i]
__global__ void k_agg_init(const float* __restrict__ H,
                           const float* __restrict__ as, const float* __restrict__ ad,
                           const float* __restrict__ m, const float* __restrict__ dn,
                           float* __restrict__ Y) {
    int idx = blockIdx.x * blockDim.x + threadIdx.x;
    if (idx >= N_NODES * D) return;
    int i = idx >> 7;
    float w = __expf(leaky(as[i] + ad[i]) - m[i]) / dn[i];
    Y[idx] = w * H[idx];
}

// Pass 3: wave per edge; lane moves a float4 of h[src], 4 fp32 atomics to Y[dst]
__global__ __launch_bounds__(256) void k_edge_agg(const int* __restrict__ src,
                                                  const int* __restrict__ dst,
                                                  const float* __restrict__ H,
                                                  const float* __restrict__ as,
                                                  const float* __restrict__ ad,
                                                  const float* __restrict__ m,
                                                  const float* __restrict__ dn,
                                                  float* __restrict__ Y, int E) {
    const int lane = threadIdx.x & 31;
    const int e = (int)((blockIdx.x * (size_t)blockDim.x + threadIdx.x) >> 5);
    if (e >= E) return;
    int s = src[e], d = dst[e];
    float v = leaky(as[s] + ad[d]);
    float w = __expf(v - m[d]) / dn[d];
    v4f h = *(const v4f*)(H + (size_t)s * D + lane * 4);
    float* y = Y + (size_t)d * D + lane * 4;
    atomicAdd(y + 0, w * h.x);
    atomicAdd(y + 1, w * h.y);
    atomicAdd(y + 2, w * h.z);
    atomicAdd(y + 3, w * h.w);
}

// Bias (+ optional ReLU), in place
__global__ void k_post(float* __restrict__ Y, const float* __restrict__ b, int relu) {
    int idx = blockIdx.x * blockDim.x + threadIdx.x;
    if (idx >= N_NODES * D) return;
    float v = Y[idx] + b[idx & (D - 1)];
    if (relu) v = fmaxf(v, 0.f);
    Y[idx] = v;
}

extern "C" void kernel_launch(void* const* d_in, const int* in_sizes, int n_in,
                              void* d_out, int out_size, void* d_ws, size_t ws_size,
                              hipStream_t stream) {
    (void)n_in; (void)out_size; (void)ws_size;
    const float* x   = (const float*)d_in[0];
    const int*   ei  = (const int*)d_in[1];
    const int    E   = in_sizes[1] / 2;
    const int*   src = ei;
    const int*   dst = ei + E;
    const float* W1 = (const float*)d_in[2];
    const float* aS1 = (const float*)d_in[3];
    const float* aD1 = (const float*)d_in[4];
    const float* b1 = (const float*)d_in[5];
    const float* W2 = (const float*)d_in[6];
    const float* aS2 = (const float*)d_in[7];
    const float* aD2 = (const float*)d_in[8];
    const float* b2 = (const float*)d_in[9];

    float* ws = (float*)d_ws;
    float* H  = ws;                               // N*D
    float* as = H + (size_t)N_NODES * D;          // N
    float* ad = as + N_NODES;                     // N
    float* m  = ad + N_NODES;                     // N
    float* dn = m + N_NODES;                      // N
    float* Y1 = dn + N_NODES;                     // N*D  (layer-1 output)
    float* Y2 = (float*)d_out;

    const int BLK = 256;
    const int gN    = (N_NODES + BLK - 1) / BLK;
    const int gNW   = (N_NODES * 32 + BLK - 1) / BLK;      // wave per node
    const int gND   = (N_NODES * D + BLK - 1) / BLK;       // thread per elem
    const int gE    = (E + BLK - 1) / BLK;                 // thread per edge
    const int gEW   = (int)(((size_t)E * 32 + BLK - 1) / BLK); // wave per edge
    const int gGemm = N_NODES / 16;                        // 50000 % 16 == 0

    // ----- layer 1 -----
    k_gemm      <<<gGemm, BLK, 0, stream>>>(x, W1, H);
    k_alpha     <<<gNW,   BLK, 0, stream>>>(H, aS1, aD1, as, ad, m);
    k_edge_max  <<<gE,    BLK, 0, stream>>>(src, dst, as, ad, m, E);
    k_denom_init<<<gN,    BLK, 0, stream>>>(as, ad, m, dn);
    k_edge_denom<<<gE,    BLK, 0, stream>>>(src, dst, as, ad, m, dn, E);
    k_agg_init  <<<gND,   BLK, 0, stream>>>(H, as, ad, m, dn, Y1);
    k_edge_agg  <<<gEW,   BLK, 0, stream>>>(src, dst, H, as, ad, m, dn, Y1, E);
    k_post      <<<gND,   BLK, 0, stream>>>(Y1, b1, /*relu=*/1);

    // ----- layer 2 -----
    k_gemm      <<<gGemm, BLK, 0, stream>>>(Y1, W2, H);
    k_alpha     <<<gNW,   BLK, 0, stream>>>(H, aS2, aD2, as, ad, m);
    k_edge_max  <<<gE,    BLK, 0, stream>>>(src, dst, as, ad, m, E);
    k_denom_init<<<gN,    BLK, 0, stream>>>(as, ad, m, dn);
    k_edge_denom<<<gE,    BLK, 0, stream>>>(src, dst, as, ad, m, dn, E);
    k_agg_init  <<<gND,   BLK, 0, stream>>>(H, as, ad, m, dn, Y2);
    k_edge_agg  <<<gEW,   BLK, 0, stream>>>(src, dst, H, as, ad, m, dn, Y2, E);
    k_post      <<<gND,   BLK, 0, stream>>>(Y2, b2, /*relu=*/0);
}